// InstLM_26096221291298
// MI455X (gfx1250) — compile-verified
//
#include <hip/hip_runtime.h>
#include <hip/hip_bf16.h>
#include <math.h>

#define RNN   64
#define TT    127      // timesteps (inst[:, :-1])
#define NGATE 256      // 4*RNN
#define NV    10001    // VOCAB+1
#define MROWS 8128     // B*TT
#define NTILE 626      // ceil(NV/16); tiles 0..624 fully in-range, 625 = tail
#define NFULL 625

typedef __attribute__((ext_vector_type(2))) float v2f;
typedef __attribute__((ext_vector_type(8))) float v8f;

// ---------------------------------------------------------------------------
// Kernel 1: embedding + LSTM recurrence. One block per batch element,
// one thread per gate row; weights held in registers, x/h via LDS.
// ---------------------------------------------------------------------------
__global__ __launch_bounds__(NGATE) void lstm_kernel(
    const int*   __restrict__ inst,     // (64,128) int32
    const float* __restrict__ embed_w,  // (NV,64)
    const float* __restrict__ w_ih,     // (256,64)
    const float* __restrict__ w_hh,     // (256,64)
    const float* __restrict__ b_ih,     // (256)
    const float* __restrict__ b_hh,     // (256)
    float*       __restrict__ H)        // (MROWS,64)
{
    const int b = blockIdx.x;
    const int j = threadIdx.x;          // gate row 0..255

    __shared__ float sh_x[RNN];
    __shared__ float sh_h[RNN];
    __shared__ float sh_g[NGATE];

    float wih[RNN], whh[RNN];
#pragma unroll
    for (int k = 0; k < RNN; ++k) {
        wih[k] = w_ih[j * RNN + k];
        whh[k] = w_hh[j * RNN + k];
    }
    const float bias = b_ih[j] + b_hh[j];

    float c = 0.0f;
    if (j < RNN) sh_h[j] = 0.0f;
    __syncthreads();

    for (int t = 0; t < TT; ++t) {
        if (j < RNN) {
            const int idx = inst[b * 128 + t];
            sh_x[j] = embed_w[(size_t)idx * RNN + j];
        }
        __syncthreads();

        float acc = bias;
#pragma unroll
        for (int k = 0; k < RNN; ++k) acc += wih[k] * sh_x[k];
#pragma unroll
        for (int k = 0; k < RNN; ++k) acc += whh[k] * sh_h[k];
        sh_g[j] = acc;
        __syncthreads();

        if (j < RNN) {
            const float gi = sh_g[j];
            const float gf = sh_g[RNN + j];
            const float gg = sh_g[2 * RNN + j];
            const float go = sh_g[3 * RNN + j];
            const float ig = 1.0f / (1.0f + expf(-gi));
            const float fg = 1.0f / (1.0f + expf(-gf));
            const float g  = tanhf(gg);
            const float og = 1.0f / (1.0f + expf(-go));
            c = fg * c + ig * g;
            const float hn = og * tanhf(c);
            sh_h[j] = hn;
            H[((size_t)b * TT + t) * RNN + j] = hn;
        }
        __syncthreads();
    }
}

// ---------------------------------------------------------------------------
// Kernel 2: logits = H @ w_out^T + b_out, fused log_softmax.
// Block = 16 rows; 4 waves stride the 626 column tiles.  WMMA f32 16x16x4.
// Pass 1 (branchless, double-buffered B): compute+store logits, sum exp(v)
// per row (logits are O(1), no max subtraction needed in f32).
// LDS sum-reduce -> 16 row LSEs.  Fence, then coalesced float4 pass 2.
// ---------------------------------------------------------------------------
__global__ __launch_bounds__(128) void logits_kernel(
    const float* __restrict__ H,      // (MROWS,64)
    const float* __restrict__ w_out,  // (NV,64)
    const float* __restrict__ b_out,  // (NV)
    float*       __restrict__ out)    // (MROWS,NV)
{
    const int m0    = blockIdx.x * 16;
    const int tid   = threadIdx.x;
    const int wave  = tid >> 5;
    const int lane  = tid & 31;
    const int lhalf = lane >> 4;   // 0: lanes 0-15, 1: lanes 16-31
    const int lmod  = lane & 15;

    __shared__ float red_s[4][8][32];
    __shared__ float sh_lse[16];

    // A fragments: A[m][k] = H[m0+m][k].  32-bit 16x4 striping:
    // lane = row lmod, K = kk*4 + 2*lhalf + {0,1}  -> one float2 load each.
    v2f a[16];
#pragma unroll
    for (int kk = 0; kk < 16; ++kk)
        a[kk] = *(const v2f*)(H + (size_t)(m0 + lmod) * RNN + kk * 4 + 2 * lhalf);

    float rs[8];
#pragma unroll
    for (int r = 0; r < 8; ++r) rs[r] = 0.0f;

    auto loadB = [&](int ctl, v2f* bf) {
        const int col = ctl * 16 + lmod;
#pragma unroll
        for (int kk = 0; kk < 16; ++kk)
            bf[kk] = *(const v2f*)(w_out + (size_t)col * RNN + kk * 4 + 2 * lhalf);
    };

    auto computeTile = [&](int ctl, const v2f* bf) {
        v8f acc = {0.f, 0.f, 0.f, 0.f, 0.f, 0.f, 0.f, 0.f};
#pragma unroll
        for (int kk = 0; kk < 16; ++kk)
            acc = __builtin_amdgcn_wmma_f32_16x16x4_f32(
                false, a[kk], false, bf[kk], (short)0, acc, false, false);
        const int   col  = ctl * 16 + lmod;
        const float bias = b_out[col];
#pragma unroll
        for (int r = 0; r < 8; ++r) {
            const float v = acc[r] + bias;   // row m0 + r + 8*lhalf, column col
            out[(size_t)(m0 + r + 8 * lhalf) * NV + col] = v;
            rs[r] += __expf(v);              // reference point 0: logits are O(1)
        }
    };

    // ---- Pass 1: full tiles, ping-pong double buffering --------------------
    v2f bfA[16], bfB[16];
    int ct = wave;
    if (ct < NFULL) loadB(ct, bfA);
    while (ct < NFULL) {
        const int ct2 = ct + 4;
        if (ct2 < NFULL) loadB(ct2, bfB);
        computeTile(ct, bfA);
        if (ct2 < NFULL) {
            const int ct3 = ct2 + 4;
            if (ct3 < NFULL) loadB(ct3, bfA);
            computeTile(ct2, bfB);
        }
        ct += 8;
    }

    // ---- Tail tile 625 (cols 10000..10015, only col 10000 valid; wave 1) ---
    if (wave == 1) {
        const int  col   = NFULL * 16 + lmod;          // 10000 + lmod
        const int  colc  = (col < NV) ? col : (NV - 1);
        const bool valid = (col < NV);
        v2f bf[16];
#pragma unroll
        for (int kk = 0; kk < 16; ++kk)
            bf[kk] = *(const v2f*)(w_out + (size_t)colc * RNN + kk * 4 + 2 * lhalf);
        v8f acc = {0.f, 0.f, 0.f, 0.f, 0.f, 0.f, 0.f, 0.f};
#pragma unroll
        for (int kk = 0; kk < 16; ++kk)
            acc = __builtin_amdgcn_wmma_f32_16x16x4_f32(
                false, a[kk], false, bf[kk], (short)0, acc, false, false);
        const float bias = b_out[colc];
#pragma unroll
        for (int r = 0; r < 8; ++r) {
            const float v = acc[r] + bias;
            if (valid) {
                out[(size_t)(m0 + r + 8 * lhalf) * NV + col] = v;
                rs[r] += __expf(v);
            }
        }
    }

    // ---- LSE reduction -----------------------------------------------------
#pragma unroll
    for (int r = 0; r < 8; ++r) red_s[wave][r][lane] = rs[r];
    __threadfence();           // flush pass-1 logit stores for cross-wave reads
    __syncthreads();

    if (tid < 16) {            // one thread per row m = tid
        const int r  = tid & 7;
        const int lh = tid >> 3;
        float S = 0.0f;
        for (int w = 0; w < 4; ++w)
            for (int l = 0; l < 16; ++l)
                S += red_s[w][r][lh * 16 + l];
        sh_lse[tid] = logf(S);
    }
    __syncthreads();

    // ---- Pass 2: coalesced float4 sweep over the 16-row stripe (L2 hits) ---
    for (int m = 0; m < 16; ++m) {
        const float l   = sh_lse[m];
        float*      row = out + (size_t)(m0 + m) * NV;
        // rows are only 4B-aligned (NV odd): peel to 16B alignment
        const int peel = (4 - ((m0 + m) & 3)) & 3;     // (m0+m)*NV mod 4 == (m0+m) mod 4
        if (tid < peel) row[tid] -= l;
        const int n4 = (NV - peel) >> 2;
        float4* row4 = (float4*)(row + peel);
        for (int i = tid; i < n4; i += 128) {
            float4 x = row4[i];
            x.x -= l; x.y -= l; x.z -= l; x.w -= l;
            row4[i] = x;
        }
        const int rem_base = peel + (n4 << 2);
        if (tid < NV - rem_base) row[rem_base + tid] -= l;
    }
}

extern "C" void kernel_launch(void* const* d_in, const int* in_sizes, int n_in,
                              void* d_out, int out_size, void* d_ws, size_t ws_size,
                              hipStream_t stream) {
    const int*   inst    = (const int*)  d_in[0];
    const float* embed_w = (const float*)d_in[1];
    const float* w_ih    = (const float*)d_in[2];
    const float* w_hh    = (const float*)d_in[3];
    const float* b_ih    = (const float*)d_in[4];
    const float* b_hh    = (const float*)d_in[5];
    const float* w_out   = (const float*)d_in[6];
    const float* b_out   = (const float*)d_in[7];
    float*       out     = (float*)d_out;
    float*       H       = (float*)d_ws;   // 8128*64*4 = 2.08 MB

    lstm_kernel<<<64, NGATE, 0, stream>>>(inst, embed_w, w_ih, w_hh,
                                          b_ih, b_hh, H);
    logits_kernel<<<MROWS / 16, 128, 0, stream>>>(H, w_out, b_out, out);
}